// MultiHeadAttention_65481071409553
// MI455X (gfx1250) — compile-verified
//
#include <hip/hip_runtime.h>

#define BB 16     // batch
#define SS 2048   // sequence
#define DD 16     // head dim (= N_WIRES)
#define WPB 8     // waves per block (wave32)

typedef __attribute__((ext_vector_type(2))) float v2f;
typedef __attribute__((ext_vector_type(4))) float v4f;
typedef __attribute__((ext_vector_type(8))) float v8f;
typedef __attribute__((ext_vector_type(4))) int   v4i;

// ---------------------------------------------------------------------------
// Kernel 1: qkv[b,s,d] = cos(x[b,s,d] + theta[d])   (2 MB -> L2 resident)
// ---------------------------------------------------------------------------
__global__ __launch_bounds__(256) void qkv_cos_kernel(const float* __restrict__ x,
                                                      const float* __restrict__ theta,
                                                      float* __restrict__ qkv) {
    int i = blockIdx.x * 256 + threadIdx.x;      // grid sized exactly to B*S*D
    qkv[i] = __cosf(x[i] + theta[i & (DD - 1)]);
}

// ---------------------------------------------------------------------------
// Kernel 2: flash attention, one wave = 16 queries, f32 WMMA 16x16x4 chained.
// Scores computed transposed: S^T(16k x 16q) so softmax reduces over regs +
// one shfl_xor(16); per-query online-softmax state is per-lane scalar.
// ---------------------------------------------------------------------------
__global__ __launch_bounds__(32 * WPB) void flash_attn_kernel(const float* __restrict__ qkv,
                                                              const int* __restrict__ mask,
                                                              float* __restrict__ out) {
    const int lane = threadIdx.x & 31;
    const int wave = threadIdx.x >> 5;
    const int lq   = lane & 15;   // query column (and generic lo-lane index)
    const int hi   = lane >> 4;   // which 16-lane half

    const int tiles_per_batch = SS / (16 * WPB);                       // 16
    const int b  = blockIdx.x / tiles_per_batch;
    const int q0 = (blockIdx.x % tiles_per_batch) * (16 * WPB) + wave * 16;

    const float* qb   = qkv + (size_t)b * SS * DD;
    const int*   mrow = mask + ((size_t)b * SS + (size_t)(q0 + lq)) * SS;

    // Q^T as B-matrix fragments (constant over the key loop).
    // bq[j].x = Q[q0+lq][4j+2*hi], .y = Q[q0+lq][4j+2*hi+1]
    v2f bq[4];
    #pragma unroll
    for (int j = 0; j < 4; ++j)
        bq[j] = *(const v2f*)(qb + (q0 + lq) * DD + 4 * j + 2 * hi);

    float m_run = -3.0e38f;   // running row max
    float l_run = 0.0f;       // running row sum
    v8f   o     = {};         // O^T accumulator (C-layout: d = g+8*hi, q = lq)

    for (int k0 = 0; k0 < SS; k0 += 16) {
        // ---- mask stream (non-temporal; this lane needs k = k0+8*hi .. +7) ----
        v4i mA = __builtin_nontemporal_load((const v4i*)(mrow + k0 + 8 * hi));
        v4i mB = __builtin_nontemporal_load((const v4i*)(mrow + k0 + 8 * hi + 4));
        __builtin_prefetch(mrow + k0 + 8 * hi + 128, 0, 0);   // ~8 tiles ahead

        // ---- K-tile as A-matrix fragments: ak[j].x = K[k0+lq][4j+2hi] ----
        v2f ak[4];
        #pragma unroll
        for (int j = 0; j < 4; ++j)
            ak[j] = *(const v2f*)(qb + (k0 + lq) * DD + 4 * j + 2 * hi);

        // ---- S^T = Ktile(16k x 16d) * Q^T(16d x 16q), chained over d ----
        v8f c = {};
        #pragma unroll
        for (int j = 0; j < 4; ++j)
            c = __builtin_amdgcn_wmma_f32_16x16x4_f32(false, ak[j], false, bq[j],
                                                      (short)0, c, false, false);

        // ---- scale + mask; element g is (q = q0+lq, k = k0+g+8*hi) ----
        float sc[8];
        #pragma unroll
        for (int g = 0; g < 8; ++g) sc[g] = c[g] * 0.25f;   // 1/sqrt(16)
        #pragma unroll
        for (int g = 0; g < 4; ++g) {
            if (mA[g] == 0) sc[g]     = -1.0e9f;
            if (mB[g] == 0) sc[g + 4] = -1.0e9f;
        }

        // ---- online softmax (k spread over 8 regs x 2 lane-halves) ----
        float tmax = sc[0];
        #pragma unroll
        for (int g = 1; g < 8; ++g) tmax = fmaxf(tmax, sc[g]);
        tmax = fmaxf(tmax, __shfl_xor(tmax, 16, 32));
        float m_new = fmaxf(m_run, tmax);
        float alpha = __expf(m_run - m_new);
        float p[8];
        float rsum = 0.0f;
        #pragma unroll
        for (int g = 0; g < 8; ++g) { p[g] = __expf(sc[g] - m_new); rsum += p[g]; }
        rsum += __shfl_xor(rsum, 16, 32);
        l_run = l_run * alpha + rsum;
        m_run = m_new;
        #pragma unroll
        for (int g = 0; g < 8; ++g) o[g] *= alpha;

        // ---- transpose P^T (C-layout) -> B-matrix fragments via lane permutes ----
        // row k lives in reg (k&7), lanes (k<8 ? 0-15 : 16-31), col q = lane&15
        v2f bp[4];
        #pragma unroll
        for (int j = 0; j < 4; ++j) {
            const int base = (j < 2) ? 4 * j : 4 * j - 8;
            const int soff = (j < 2) ? 0 : 16;
            float x0 = __shfl(p[base + 0], lq + soff, 32);
            float x1 = __shfl(p[base + 1], lq + soff, 32);
            float x2 = __shfl(p[base + 2], lq + soff, 32);
            float x3 = __shfl(p[base + 3], lq + soff, 32);
            bp[j].x = hi ? x2 : x0;
            bp[j].y = hi ? x3 : x1;
        }

        // ---- V^T as A-matrix fragments: av[j].x = V[k0+4j+2hi][d=lq] ----
        v2f av[4];
        #pragma unroll
        for (int j = 0; j < 4; ++j) {
            av[j].x = qb[(k0 + 4 * j + 2 * hi) * DD + lq];
            av[j].y = qb[(k0 + 4 * j + 2 * hi + 1) * DD + lq];
        }

        // ---- O^T += V^T(16d x 16k) * P^T(16k x 16q) ----
        #pragma unroll
        for (int j = 0; j < 4; ++j)
            o = __builtin_amdgcn_wmma_f32_16x16x4_f32(false, av[j], false, bp[j],
                                                      (short)0, o, false, false);
    }

    // ---- epilogue: out[b][q0+lq][g+8*hi] = O^T / l ----
    const float linv = 1.0f / l_run;
    float* orow = out + ((size_t)b * SS + (size_t)(q0 + lq)) * DD + 8 * hi;
    v4f o0 = { o[0] * linv, o[1] * linv, o[2] * linv, o[3] * linv };
    v4f o1 = { o[4] * linv, o[5] * linv, o[6] * linv, o[7] * linv };
    *(v4f*)(orow)     = o0;
    *(v4f*)(orow + 4) = o1;
}

// ---------------------------------------------------------------------------
extern "C" void kernel_launch(void* const* d_in, const int* in_sizes, int n_in,
                              void* d_out, int out_size, void* d_ws, size_t ws_size,
                              hipStream_t stream) {
    (void)in_sizes; (void)n_in; (void)out_size; (void)ws_size;
    const float* x     = (const float*)d_in[0];
    const int*   mask  = (const int*)d_in[1];
    const float* theta = (const float*)d_in[2];
    float*       out   = (float*)d_out;
    float*       qkv   = (float*)d_ws;   // B*S*D*4 = 2 MB scratch

    qkv_cos_kernel<<<(BB * SS * DD) / 256, 256, 0, stream>>>(x, theta, qkv);
    flash_attn_kernel<<<BB * (SS / (16 * WPB)), 32 * WPB, 0, stream>>>(qkv, mask, out);
}